// CapsuleLayer_35502199669312
// MI455X (gfx1250) — compile-verified
//
#include <hip/hip_runtime.h>
#include <hip/hip_bf16.h>

// Capsule routing for MI455X (gfx1250, wave32), WMMA f32, multi-kernel.
//
//   inputs : [512, 144, 8]  f32      W : [32, 144, 16, 8] f32
//   out    : [512, 32, 16]  f32
//
// inputs_hat (151 MB) is never materialized: u_hat tiles are recomputed with
// V_WMMA_F32_16X16X4_F32 wherever routing needs them. Routing logits are
// linear in the running sum of squashed outputs (vsum), so per-(b,nc,c) logit
// state collapses to vsum [512,32,16] (1 MB). The only cross-capsule coupling
// is the softmax denominator, so the 3 routing iterations become 7
// stream-ordered launches (stream order = global barrier), exchanging vsum +
// softmax numerators (10.5 MB) through d_ws — fully L2-resident (192 MB L2).
// Critical path is shortened by splitting the c-reduction: 2-way inside the
// S workgroup (LDS combine), 4-way across T workgroups (1024 WGs).

#define B_TOT 512
#define NC    32
#define DC    16
#define CIN   144
#define DIN   8
#define EPSV  1e-7f

typedef float v2f __attribute__((ext_vector_type(2)));
typedef float v8f __attribute__((ext_vector_type(8)));

// squash per C-tile row (batch): norm over j = over the 16 lanes of a half.
__device__ __forceinline__ v8f squash_rows(v8f s) {
  v8f v;
  #pragma unroll
  for (int vv = 0; vv < 8; ++vv) {
    float n2 = s[vv] * s[vv];
    n2 += __shfl_xor(n2, 1);
    n2 += __shfl_xor(n2, 2);
    n2 += __shfl_xor(n2, 4);
    n2 += __shfl_xor(n2, 8);
    const float f = n2 / ((1.0f + n2) * sqrtf(n2 + EPSV));
    v[vv] = s[vv] * f;
  }
  return v;
}

// ---------------------------------------------------------------------------
// S kernel: s[b,j] = sum_c coef[b,c] * u_hat[b,c,j]  (coef folded into A rows
// => pure chained-accumulator WMMA), then squash.
// 8 waves = 4 nc x 2 c-halves; partial accumulators combined through LDS.
// MODE 0: coef = 1/32 (softmax of zeros), vsum  = v      (iteration 0)
// MODE 1: coef from e_ws,                 vsum += v      (iteration 1)
// MODE 2: coef from e_ws,                 out   = v      (iteration 2)
// grid (32 batch-tiles, 8 nc-quads), 256 threads.
// ---------------------------------------------------------------------------
template <int MODE>
__global__ __launch_bounds__(256)
void caps_s_kernel(const float* __restrict__ x, const float* __restrict__ W,
                   const float* __restrict__ coef, float* __restrict__ vsum,
                   float* __restrict__ out)
{
  __shared__ float sbuf[4 * 256];              // partial s per nc-quad (4 KB)

  const int lane  = threadIdx.x & 31;
  const int wave  = threadIdx.x >> 5;
  const int ncq   = wave & 3;                  // which nc of the quad
  const int chalf = wave >> 2;                 // which half of the c range
  const int half  = lane >> 4;
  const int m16   = lane & 15;
  const int b0    = blockIdx.x * 16;
  const int nc    = blockIdx.y * 4 + ncq;
  const int c0    = chalf * (CIN / 2);

  // A: a[v] = A[m16][2*half + v + k0]   B: b[v] = B[2*half + v + k0][m16]
  const int xbase = ((b0 + m16) * CIN) * DIN + 2 * half;           // + c*8
  const int wbase = ((nc * CIN) * DC + m16) * DIN + 2 * half;      // + c*128

  v8f s = {};
  if (MODE == 0) {
    #pragma unroll 4
    for (int c = c0; c < c0 + CIN / 2; ++c) {
      v2f a0 = *(const v2f*)(x + xbase + c * DIN);
      v2f a1 = *(const v2f*)(x + xbase + c * DIN + 4);
      v2f w0 = *(const v2f*)(W + wbase + c * (DC * DIN));
      v2f w1 = *(const v2f*)(W + wbase + c * (DC * DIN) + 4);
      s = __builtin_amdgcn_wmma_f32_16x16x4_f32(false, a0, false, w0, (short)0, s, false, false);
      s = __builtin_amdgcn_wmma_f32_16x16x4_f32(false, a1, false, w1, (short)0, s, false, false);
    }
  } else {
    const int cbase = ((b0 + m16) * NC + nc) * CIN;
    #pragma unroll 4
    for (int c = c0; c < c0 + CIN / 2; ++c) {
      const float cf = coef[cbase + c];
      v2f a0 = *(const v2f*)(x + xbase + c * DIN)     * cf;
      v2f a1 = *(const v2f*)(x + xbase + c * DIN + 4) * cf;
      v2f w0 = *(const v2f*)(W + wbase + c * (DC * DIN));
      v2f w1 = *(const v2f*)(W + wbase + c * (DC * DIN) + 4);
      s = __builtin_amdgcn_wmma_f32_16x16x4_f32(false, a0, false, w0, (short)0, s, false, false);
      s = __builtin_amdgcn_wmma_f32_16x16x4_f32(false, a1, false, w1, (short)0, s, false, false);
    }
  }

  // combine the two c-halves (identical lane indexing on both sides)
  if (chalf == 1) {
    #pragma unroll
    for (int vv = 0; vv < 8; ++vv)
      sbuf[ncq * 256 + vv * 32 + lane] = s[vv];
  }
  __syncthreads();
  if (chalf == 1) return;

  #pragma unroll
  for (int vv = 0; vv < 8; ++vv)
    s[vv] += sbuf[ncq * 256 + vv * 32 + lane];
  if (MODE == 0) s *= (1.0f / NC);             // exact uniform softmax coef

  const v8f v = squash_rows(s);

  #pragma unroll
  for (int vv = 0; vv < 8; ++vv) {
    const int a = ((b0 + vv + 8 * half) * NC + nc) * DC + m16;
    if (MODE == 2)      out[a]  = v[vv];
    else if (MODE == 0) vsum[a] = v[vv];
    else                vsum[a] = vsum[a] + v[vv];
  }
}

// ---------------------------------------------------------------------------
// T kernel: logits b[b,nc,c] = <vsum[b,nc,:], u_hat[b,nc,c,:]> ; e = exp(b).
// u_hat tile per c via 2 WMMAs (same operands as S, coef=1); row-wise dot
// against vsum via a register-folding tree (15 shfl + 15 add + 7 cndmask for
// all 16 rows), leaving the 8 row-sums lane-packed => a single v_exp_f32.
// grid (32 batch-tiles, 8 nc-quads, 4 c-quarters), 128 threads => 1024 WGs.
// ---------------------------------------------------------------------------
__global__ __launch_bounds__(128)
void caps_t_kernel(const float* __restrict__ x, const float* __restrict__ W,
                   const float* __restrict__ vsum, float* __restrict__ e_ws)
{
  const int lane = threadIdx.x & 31;
  const int wave = threadIdx.x >> 5;
  const int half = lane >> 4;
  const int m16  = lane & 15;
  const int b0   = blockIdx.x * 16;
  const int nc   = blockIdx.y * 4 + wave;
  const int c0   = blockIdx.z * (CIN / 4);

  const int xbase = ((b0 + m16) * CIN) * DIN + 2 * half;
  const int wbase = ((nc * CIN) * DC + m16) * DIN + 2 * half;

  // vsum in C-layout: element [m = vv+8*half][n=j = m16]
  v8f vsc;
  #pragma unroll
  for (int vv = 0; vv < 8; ++vv)
    vsc[vv] = vsum[((b0 + vv + 8 * half) * NC + nc) * DC + m16];

  const bool s3 = (lane & 8) != 0;
  const bool s2 = (lane & 4) != 0;
  const bool s1 = (lane & 2) != 0;
  // after folds on lane bits 3,2,1 the full sum of row 'row' lands here:
  const int row   = 4 * ((lane >> 1) & 1) + 2 * ((lane >> 2) & 1) + ((lane >> 3) & 1) + 8 * half;
  const int ebase = ((b0 + row) * NC + nc) * CIN;

  #pragma unroll 2
  for (int c = c0; c < c0 + CIN / 4; ++c) {
    v2f a0 = *(const v2f*)(x + xbase + c * DIN);
    v2f a1 = *(const v2f*)(x + xbase + c * DIN + 4);
    v2f w0 = *(const v2f*)(W + wbase + c * (DC * DIN));
    v2f w1 = *(const v2f*)(W + wbase + c * (DC * DIN) + 4);
    v8f u = {};
    u = __builtin_amdgcn_wmma_f32_16x16x4_f32(false, a0, false, w0, (short)0, u, false, false);
    u = __builtin_amdgcn_wmma_f32_16x16x4_f32(false, a1, false, w1, (short)0, u, false, false);

    float p[8];
    #pragma unroll
    for (int vv = 0; vv < 8; ++vv) {
      p[vv] = u[vv] * vsc[vv];          // elementwise vs vsum (same C-layout)
      p[vv] += __shfl_xor(p[vv], 8);    // pair-sums over lane bit 3
    }
    float q[4];
    #pragma unroll
    for (int k = 0; k < 4; ++k) {       // fold rows by bit3, reduce bit2
      q[k] = s3 ? p[2 * k + 1] : p[2 * k];
      q[k] += __shfl_xor(q[k], 4);
    }
    float r[2];
    #pragma unroll
    for (int j = 0; j < 2; ++j) {       // fold by bit2, reduce bit1
      r[j] = s2 ? q[2 * j + 1] : q[2 * j];
      r[j] += __shfl_xor(r[j], 2);
    }
    float uu = s1 ? r[1] : r[0];        // fold by bit1, reduce bit0
    uu += __shfl_xor(uu, 1);

    const float e = __expf(uu);         // one exp covers all 8 rows (lanes)
    if (!(lane & 1))
      e_ws[ebase + c] = e;
  }
}

// ---------------------------------------------------------------------------
// Softmax normalization over nc, in place on e_ws. grid 288 x 256 threads.
// ---------------------------------------------------------------------------
__global__ __launch_bounds__(256)
void caps_norm_kernel(float* __restrict__ e_ws)
{
  const int idx = blockIdx.x * 256 + threadIdx.x;   // (b, c) pair
  if (idx >= B_TOT * CIN) return;
  const int b = idx / CIN, c = idx - b * CIN;
  const int base = (b * NC) * CIN + c;
  float d = 0.0f;
  #pragma unroll 4
  for (int n = 0; n < NC; ++n) d += e_ws[base + n * CIN];
  const float inv = 1.0f / d;
  #pragma unroll 4
  for (int n = 0; n < NC; ++n) e_ws[base + n * CIN] *= inv;
}

// ---------------------------------------------------------------------------
// Fallback (ws too small): fully fused single kernel, LDS-resident softmax
// state. 32 WGs x 8 waves, 296 KB dynamic LDS.
// ---------------------------------------------------------------------------
#define E_ALL_FLOATS   (NC * CIN * 16)
#define SCRATCH_FLOATS (8 * 16 * 16)
#define SMEM_BYTES ((E_ALL_FLOATS + SCRATCH_FLOATS) * sizeof(float))

__global__ __launch_bounds__(256)
void caps_fused_kernel(const float* __restrict__ x, const float* __restrict__ W,
                       float* __restrict__ out)
{
  extern __shared__ float smem[];
  float* e_all   = smem;
  float* scratch = smem + E_ALL_FLOATS;

  const int tid  = threadIdx.x;
  const int wave = tid >> 5, lane = tid & 31;
  const int half = lane >> 4, m16 = lane & 15;
  const int b0   = blockIdx.x * 16;

  v2f vsum[4][4];
  #pragma unroll
  for (int q = 0; q < 4; ++q)
    #pragma unroll
    for (int ch = 0; ch < 4; ++ch) { vsum[q][ch].x = 0.0f; vsum[q][ch].y = 0.0f; }

  for (int t = 0; t < 3; ++t) {
    if (t > 0) {
      for (int q = 0; q < 4; ++q) {
        const int nc = wave * 4 + q;
        for (int c = 0; c < CIN; ++c) {
          v8f T = {};
          #pragma unroll
          for (int ch = 0; ch < 4; ++ch) {
            const int j0 = ch * 4 + 2 * half;
            v2f bb; bb.x = 0.0f; bb.y = 0.0f;
            if (m16 < DIN) {
              const float* wp = W + ((nc * CIN + c) * DC + j0) * DIN + m16;
              bb.x = wp[0]; bb.y = wp[DIN];
            }
            T = __builtin_amdgcn_wmma_f32_16x16x4_f32(false, vsum[q][ch], false, bb, (short)0, T, false, false);
          }
          #pragma unroll
          for (int vv = 0; vv < 8; ++vv) {
            const int m = vv + 8 * half;
            float p = 0.0f;
            if (m16 < DIN)
              p = T[vv] * x[((b0 + m) * CIN + c) * DIN + m16];
            p += __shfl_xor(p, 1); p += __shfl_xor(p, 2);
            p += __shfl_xor(p, 4); p += __shfl_xor(p, 8);
            if (m16 == 0) e_all[(nc * CIN + c) * 16 + m] = __expf(p);
          }
        }
      }
      __syncthreads();
      for (int idx = tid; idx < CIN * 16; idx += 256) {
        const int c = idx >> 4, b = idx & 15;
        float d = 0.0f;
        for (int n = 0; n < NC; ++n) d += e_all[(n * CIN + c) * 16 + b];
        const float inv = 1.0f / d;
        for (int n = 0; n < NC; ++n) e_all[(n * CIN + c) * 16 + b] *= inv;
      }
      __syncthreads();
    }
    for (int q = 0; q < 4; ++q) {
      const int nc = wave * 4 + q;
      v8f s = {};
      for (int c = 0; c < CIN; ++c) {
        const float cf = (t == 0) ? (1.0f / NC) : e_all[(nc * CIN + c) * 16 + m16];
        const float* xp = x + ((b0 + m16) * CIN + c) * DIN + 2 * half;
        v2f a0 = *(const v2f*)(xp)     * cf;
        v2f a1 = *(const v2f*)(xp + 4) * cf;
        const float* wp = W + ((nc * CIN + c) * DC + m16) * DIN + 2 * half;
        v2f w0 = *(const v2f*)(wp);
        v2f w1 = *(const v2f*)(wp + 4);
        s = __builtin_amdgcn_wmma_f32_16x16x4_f32(false, a0, false, w0, (short)0, s, false, false);
        s = __builtin_amdgcn_wmma_f32_16x16x4_f32(false, a1, false, w1, (short)0, s, false, false);
      }
      const v8f v = squash_rows(s);
      if (t == 2) {
        #pragma unroll
        for (int vv = 0; vv < 8; ++vv)
          out[(b0 + vv + 8 * half) * (NC * DC) + nc * DC + m16] = v[vv];
      } else {
        float* sc = scratch + wave * 256;
        #pragma unroll
        for (int vv = 0; vv < 8; ++vv)
          sc[(vv + 8 * half) * 16 + m16] = v[vv];
        __syncthreads();
        #pragma unroll
        for (int ch = 0; ch < 4; ++ch) {
          const int j0 = ch * 4 + 2 * half;
          vsum[q][ch].x += sc[m16 * 16 + j0];
          vsum[q][ch].y += sc[m16 * 16 + j0 + 1];
        }
        __syncthreads();
      }
    }
    __syncthreads();
  }
}

// ---------------------------------------------------------------------------
extern "C" void kernel_launch(void* const* d_in, const int* in_sizes, int n_in,
                              void* d_out, int out_size, void* d_ws, size_t ws_size,
                              hipStream_t stream) {
  (void)in_sizes; (void)n_in; (void)out_size;
  const float* x = (const float*)d_in[0];   // [512,144,8]
  const float* W = (const float*)d_in[1];   // [32,144,16,8]
  float* out     = (float*)d_out;           // [512,32,16]

  const size_t VSUM_N = (size_t)B_TOT * NC * DC;    //   256 K floats (1 MB)
  const size_t E_N    = (size_t)B_TOT * NC * CIN;   //  2.36 M floats (9.4 MB)
  const size_t WS_NEED = (VSUM_N + E_N) * sizeof(float);

  if (ws_size >= WS_NEED) {
    float* vsum = (float*)d_ws;
    float* e_ws = vsum + VSUM_N;
    dim3 gS(B_TOT / 16, NC / 4);        // 256 WGs x 8 waves
    dim3 gT(B_TOT / 16, NC / 4, 4);     // 1024 WGs x 4 waves
    dim3 gN((B_TOT * CIN + 255) / 256); // 288 WGs
    // stream order provides the cross-capsule softmax barriers
    caps_s_kernel<0><<<gS, 256, 0, stream>>>(x, W, nullptr, vsum, nullptr);
    caps_t_kernel   <<<gT, 128, 0, stream>>>(x, W, vsum, e_ws);
    caps_norm_kernel<<<gN, 256, 0, stream>>>(e_ws);
    caps_s_kernel<1><<<gS, 256, 0, stream>>>(x, W, e_ws, vsum, nullptr);
    caps_t_kernel   <<<gT, 128, 0, stream>>>(x, W, vsum, e_ws);
    caps_norm_kernel<<<gN, 256, 0, stream>>>(e_ws);
    caps_s_kernel<2><<<gS, 256, 0, stream>>>(x, W, e_ws, nullptr, out);
  } else {
    caps_fused_kernel<<<B_TOT / 16, 256, SMEM_BYTES, stream>>>(x, W, out);
  }
}